// MeshLaplacianLoss_10007273800421
// MI455X (gfx1250) — compile-verified
//
#include <hip/hip_runtime.h>

typedef __attribute__((ext_vector_type(2))) float v2f;
typedef __attribute__((ext_vector_type(8))) float v8f;

#define NVERT 9216   // 96*96
#define BATCH 8
#define NCOLS 32     // 24 live columns (8 batches * 3 coords) padded to 32

// ---------------------------------------------------------------------------
// Kernel 1: pack verts [B, N, 3] fp32 into pair-interleaved Xp:
//   Xp[(k>>1)*64 + col*2 + (k&1)] = X[k, col],  X[k, col] = verts[b, k, c]
//   with col = b*3 + c for col < 24, zero for col in [24,32).
// This makes the WMMA B-operand a single contiguous float2 per lane.
// ---------------------------------------------------------------------------
__global__ void pack_verts_kernel(const float* __restrict__ verts,
                                  float* __restrict__ Xp) {
    int idx = blockIdx.x * blockDim.x + threadIdx.x;   // 0 .. N*32-1
    int k   = idx >> 5;
    int col = idx & 31;
    float v = 0.0f;
    if (col < 24) {
        int b = col / 3;
        int c = col - b * 3;
        v = verts[(size_t)b * NVERT * 3 + (size_t)k * 3 + c];
    }
    Xp[(size_t)(k >> 1) * 64 + col * 2 + (k & 1)] = v;
}

// ---------------------------------------------------------------------------
// Kernel 2: one workgroup per 16-row tile of L (576 blocks, 256 thr = 8 waves).
// Wave w accumulates LV over K-chunk [w*1152, (w+1)*1152) with
// v_wmma_f32_16x16x4_f32, two N-tiles (cols 0-15, 16-31).
// Pointer-increment addressing: all loads in the unrolled body use immediate
// offsets; two pointer adds per 32-K body. One global_prefetch_b8 per body
// keeps the HBM stream of L ahead of WMMA consumption.
// ---------------------------------------------------------------------------
__global__ __launch_bounds__(256) void lap_wmma_kernel(
        const float* __restrict__ L,
        const float* __restrict__ Xp,
        float* __restrict__ partials) {
    __shared__ float red[8 * 512];    // 8 waves x (2 tiles x 8 vgpr x 32 lanes)
    __shared__ float red2[256];

    const int tid  = threadIdx.x;
    const int lane = tid & 31;
    const int wave = tid >> 5;
    const int r0   = blockIdx.x * 16;

    // A layout (f32 16x4): lane l<16 -> row l, K={0,1}; lane l+16 -> row l, K={2,3}
    const int m  = lane & 15;
    const int hi = lane >> 4;                       // 0 or 1
    const size_t rowBase = (size_t)(r0 + m) * NVERT;

    v8f acc0 = {0.f,0.f,0.f,0.f,0.f,0.f,0.f,0.f};
    v8f acc1 = {0.f,0.f,0.f,0.f,0.f,0.f,0.f,0.f};

    const int kChunk = NVERT / 8;                   // 1152 K per wave
    const int k0 = wave * kChunk;

    // Per-lane streaming pointers.
    const float* ap = L + rowBase + k0 + hi * 2;                    // A stream
    const float* bp = Xp + ((size_t)(k0 >> 1) + hi) * 64 + m * 2;   // B stream

    for (int ob = 0; ob < kChunk / 32; ++ob) {      // 36 bodies of 32 K
        // Prefetch the L stream 4KB ahead (speculative; tail overrun dropped).
        __builtin_prefetch((const void*)(ap + 1024), 0, 1);
#pragma unroll
        for (int u = 0; u < 8; ++u) {
            v2f a  = *(const v2f*)(ap + u * 4);          // {L[r,kb], L[r,kb+1]}
            v2f b0 = *(const v2f*)(bp + u * 128);        // cols 0..15
            v2f b1 = *(const v2f*)(bp + u * 128 + 32);   // cols 16..31
            acc0 = __builtin_amdgcn_wmma_f32_16x16x4_f32(
                false, a, false, b0, (short)0, acc0, false, false);
            acc1 = __builtin_amdgcn_wmma_f32_16x16x4_f32(
                false, a, false, b1, (short)0, acc1, false, false);
        }
        ap += 32;          // 32 K of L per body
        bp += 1024;        // 16 pair-rows * 64 floats per body
    }

    // Per-wave C tiles to LDS (layout-agnostic: all waves share the same
    // lane/vgpr -> (m,n) mapping, so elementwise cross-wave sums are exact).
#pragma unroll
    for (int i = 0; i < 8; ++i) {
        red[wave * 512 +       i * 32 + lane] = acc0[i];
        red[wave * 512 + 256 + i * 32 + lane] = acc1[i];
    }
    __syncthreads();

    // Sum each of the 512 tile elements across the 8 waves, square, accumulate.
    float sq = 0.0f;
#pragma unroll
    for (int e = tid; e < 512; e += 256) {
        float s = 0.0f;
#pragma unroll
        for (int w = 0; w < 8; ++w) s += red[w * 512 + e];
        sq += s * s;
    }
    red2[tid] = sq;
    __syncthreads();
    for (int off = 128; off > 0; off >>= 1) {
        if (tid < off) red2[tid] += red2[tid + off];
        __syncthreads();
    }
    if (tid == 0) partials[blockIdx.x] = red2[0];
}

// ---------------------------------------------------------------------------
// Kernel 3: deterministic final reduction of 576 block partials -> scalar / B.
// ---------------------------------------------------------------------------
__global__ void final_reduce_kernel(const float* __restrict__ partials,
                                    float* __restrict__ out) {
    __shared__ float s[256];
    int tid = threadIdx.x;
    float v = 0.0f;
    for (int i = tid; i < NVERT / 16; i += 256) v += partials[i];
    s[tid] = v;
    __syncthreads();
    for (int off = 128; off > 0; off >>= 1) {
        if (tid < off) s[tid] += s[tid + off];
        __syncthreads();
    }
    if (tid == 0) out[0] = s[0] / (float)BATCH;
}

extern "C" void kernel_launch(void* const* d_in, const int* in_sizes, int n_in,
                              void* d_out, int out_size, void* d_ws, size_t ws_size,
                              hipStream_t stream) {
    const float* L     = (const float*)d_in[0];   // laplacian [N,N]
    const float* verts = (const float*)d_in[1];   // verts [B,N,3]

    float* Xp       = (float*)d_ws;               // N*32 floats (1.18 MB)
    float* partials = Xp + (size_t)NVERT * NCOLS; // 576 floats

    pack_verts_kernel<<<(NVERT * NCOLS) / 256, 256, 0, stream>>>(verts, Xp);
    lap_wmma_kernel<<<NVERT / 16, 256, 0, stream>>>(L, Xp, partials);
    final_reduce_kernel<<<1, 256, 0, stream>>>(partials, (float*)d_out);
}